// MPS_Ring_79267916414994
// MI455X (gfx1250) — compile-verified
//
#include <hip/hip_runtime.h>

typedef __attribute__((ext_vector_type(2))) float v2f;
typedef __attribute__((ext_vector_type(8))) float v8f;

#define NUMS   784
#define VD     20
#define PD     2
#define CD     10
#define PAD    32            // 20 padded up to 2x 16x16 WMMA tiles (M/N); K needs only 20
#define KT     5             // k-tiles of 4: 5*4 = 20 = VD exactly (tiles 5..7 are all-zero)
#define LCHAIN (NUMS - 1)    // 783 chain matrices

__launch_bounds__(32)
__global__ void mps_chain_wmma_kernel(const float* __restrict__ input,
                                      const float* __restrict__ left,
                                      const float* __restrict__ middle,
                                      float* __restrict__ out) {
    __shared__ float sP[PAD * PAD];   // running product P, row-major; rows/cols >=20 stay 0
    __shared__ float sMt[PAD * PAD];  // step matrix transposed: sMt[j*PAD+i] = M[i][j]
    __shared__ float sX[NUMS * PD];   // this batch's input row

    const int b    = blockIdx.x;
    const int lane = threadIdx.x;     // wave32: 0..31
    const int l16  = lane & 15;
    const int half = lane >> 4;       // 0: lanes 0-15, 1: lanes 16-31

    // Preload input_data[b, :, :] (1568 floats) into LDS.
    for (int e = lane; e < NUMS * PD; e += 32)
        sX[e] = input[(size_t)b * (NUMS * PD) + e];

    // Init P = eye(20) zero-padded; Mt = 0 (padding regions stay zero forever).
    for (int e = lane; e < PAD * PAD; e += 32) {
        int i = e >> 5, j = e & 31;
        sP[e]  = (i == j && i < VD) ? 1.0f : 0.0f;
        sMt[e] = 0.0f;
    }
    __syncthreads();

    for (int l = 0; l < LCHAIN; ++l) {
        const float x0 = sX[l * PD + 0];
        const float x1 = sX[l * PD + 1];
        const float* Lp = left + (size_t)l * (VD * PD * VD);
        if (l + 1 < LCHAIN)  // speculative prefetch of next chain tile -> global_prefetch_b8
            __builtin_prefetch(Lp + VD * PD * VD, 0, 1);

        // Build M^T in LDS: sMt[j*PAD+i] = x0*left[l,i,0,j] + x1*left[l,i,1,j]
        for (int e = lane; e < VD * VD; e += 32) {
            int i = e / VD, j = e - i * VD;
            float m = x0 * Lp[i * (PD * VD) + j] + x1 * Lp[i * (PD * VD) + VD + j];
            sMt[j * PAD + i] = m;
        }
        __syncthreads();

        // P_new = P @ M via 2x2 tiles of V_WMMA_F32_16X16X4_F32, K = 5 steps of 4 (= 20).
        v8f acc00 = {}, acc01 = {}, acc10 = {}, acc11 = {};
        const int rowA0 = (0 * 16 + l16) * PAD;   // A tile rows (M on lanes)
        const int rowA1 = (1 * 16 + l16) * PAD;
        const int colB0 = (0 * 16 + l16) * PAD;   // B read from Mt rows (N on lanes)
        const int colB1 = (1 * 16 + l16) * PAD;
#pragma unroll
        for (int kt = 0; kt < KT; ++kt) {
            const int k = kt * 4 + half * 2;      // VGPR0=K=k, VGPR1=K=k+1 per ISA layout
            v2f a0 = *(const v2f*)&sP [rowA0 + k];
            v2f a1 = *(const v2f*)&sP [rowA1 + k];
            v2f b0 = *(const v2f*)&sMt[colB0 + k];
            v2f b1 = *(const v2f*)&sMt[colB1 + k];
            acc00 = __builtin_amdgcn_wmma_f32_16x16x4_f32(false, a0, false, b0, (short)0, acc00, false, false);
            acc01 = __builtin_amdgcn_wmma_f32_16x16x4_f32(false, a0, false, b1, (short)0, acc01, false, false);
            acc10 = __builtin_amdgcn_wmma_f32_16x16x4_f32(false, a1, false, b0, (short)0, acc10, false, false);
            acc11 = __builtin_amdgcn_wmma_f32_16x16x4_f32(false, a1, false, b1, (short)0, acc11, false, false);
        }
        __syncthreads();  // all reads of sP/sMt done before overwriting sP / rebuilding sMt

        // Scatter D (C/D layout: VGPR r -> row r + 8*half) back into sP.
        // Row-0 tiles: all 16 rows valid.
#pragma unroll
        for (int r = 0; r < 8; ++r) {
            const int rr = r + 8 * half;
            sP[rr * PAD + l16]      = acc00[r];
            sP[rr * PAD + 16 + l16] = acc01[r];
        }
        // Row-1 tiles: only rows 16..19 are nonzero -> VGPRs r=0..3 of half==0 lanes.
        if (half == 0) {
#pragma unroll
            for (int r = 0; r < 4; ++r) {
                sP[(r + 16) * PAD + l16]      = acc10[r];
                sP[(r + 16) * PAD + 16 + l16] = acc11[r];
            }
        }
        __syncthreads();
    }

    // out[b,c] = sum_{i,j} P[i][j] * (x0*middle[c,j,0,i] + x1*middle[c,j,1,i])
    const float x0 = sX[(NUMS - 1) * PD + 0];
    const float x1 = sX[(NUMS - 1) * PD + 1];
    for (int c = 0; c < CD; ++c) {
        float partial = 0.0f;
        for (int e = lane; e < VD * VD; e += 32) {
            int i = e / VD, j = e - i * VD;
            const float* Mp = middle + (size_t)c * (VD * PD * VD) + j * (PD * VD);
            partial += sP[i * PAD + j] * (x0 * Mp[i] + x1 * Mp[VD + i]);
        }
#pragma unroll
        for (int off = 16; off > 0; off >>= 1)
            partial += __shfl_down(partial, off, 32);
        if (lane == 0)
            out[b * CD + c] = partial;
    }
}

extern "C" void kernel_launch(void* const* d_in, const int* in_sizes, int n_in,
                              void* d_out, int out_size, void* d_ws, size_t ws_size,
                              hipStream_t stream) {
    (void)in_sizes; (void)n_in; (void)d_ws; (void)ws_size; (void)out_size;
    const float* input  = (const float*)d_in[0];  // (1024, 784, 2) f32
    const float* left   = (const float*)d_in[1];  // (783, 20, 2, 20) f32
    const float* middle = (const float*)d_in[2];  // (10, 20, 2, 20) f32
    float* out = (float*)d_out;                   // (1024, 10) f32

    mps_chain_wmma_kernel<<<1024, 32, 0, stream>>>(input, left, middle, out);
}